// AuxIVA_T_ISS_73108933313037
// MI455X (gfx1250) — compile-verified
//
#include <hip/hip_runtime.h>

#define BATCH 4
#define NCH 4
#define NF 257
#define NN 2000
#define NITER 3
#define PTOT 3   /* N_TAPS + N_DELAY */
#define NTAPS 2
#define NSPLIT 5            /* frame-range split for the reduce kernel */
#define NNCHUNK (NN / NSPLIT) /* 400, multiple of 4 */
#define EPSV 1e-3f
#define EPSM 1e-5f

#define FN (NF * NN)            /* 514000  */
#define CFN (NCH * FN)          /* 2056000 */
#define TOTAL (BATCH * CFN)     /* 8224000 */
#define BCF (BATCH * NCH * NF)  /* 4112    */
#define BCN (BATCH * NCH * NN)  /* 32000   */
#define WTOT (BATCH * NCH * NF * NCH) /* 16448 */

typedef float v2f __attribute__((ext_vector_type(2)));
typedef float v8f __attribute__((ext_vector_type(8)));

// ---------------- init ----------------
__global__ void k_init_copy(const float* __restrict__ inr, const float* __restrict__ ini,
                            float* __restrict__ xr, float* __restrict__ xi) {
  int i = blockIdx.x * blockDim.x + threadIdx.x;
  if (i < TOTAL) { xr[i] = inr[i]; xi[i] = ini[i]; }
}

__global__ void k_init_W(float* __restrict__ wr, float* __restrict__ wi) {
  int i = blockIdx.x * blockDim.x + threadIdx.x;
  if (i < WTOT) {
    int d = i % NCH;
    int c = (i / (NF * NCH)) % NCH;
    wr[i] = (c == d) ? 1.f : 0.f;
    wi[i] = 0.f;
  }
}

// -------- per-(b,c,frame) sum over freq of |X|^2 --------
__global__ void k_colsum(const float* __restrict__ xr, const float* __restrict__ xi,
                         float* __restrict__ s) {
  int n = blockIdx.x * blockDim.x + threadIdx.x;
  int c = blockIdx.y, b = blockIdx.z;
  if (n >= NN) return;
  size_t base = (size_t)(b * NCH + c) * FN + n;
  float acc = 0.f;
  for (int f = 0; f < NF; ++f) {
    float a = xr[base + (size_t)f * NN];
    float bb = xi[base + (size_t)f * NN];
    acc += a * a + bb * bb;
  }
  s[(b * NCH + c) * NN + n] = acc;
}

// -------- g (mean |X|^2 over f,n) + Laplace weights * g --------
__global__ void k_gweights(const float* __restrict__ s, float* __restrict__ wgt,
                           float* __restrict__ g, float* __restrict__ rg) {
  int bc = blockIdx.x;
  int tid = threadIdx.x;
  __shared__ float red[256];
  __shared__ float gsh;
  const float* sp = s + bc * NN;
  float acc = 0.f;
  for (int n = tid; n < NN; n += 256) acc += sp[n];
  red[tid] = acc;
  __syncthreads();
  for (int w = 128; w > 0; w >>= 1) {
    if (tid < w) red[tid] += red[tid + w];
    __syncthreads();
  }
  if (tid == 0) {
    float gv = fmaxf(red[0] / (float)(NF * NN), EPSV);
    g[bc] = gv;
    rg[bc] = 1.f / sqrtf(gv);  // sqrt(g) >= sqrt(EPS) > EPS, so plain divide
    gsh = gv;
  }
  __syncthreads();
  float gv = gsh;
  for (int n = tid; n < NN; n += 256)
    wgt[bc * NN + n] = gv / fmaxf(2.f * sqrtf(sp[n]), EPSM);
}

__global__ void k_scaleX(float* __restrict__ xr, float* __restrict__ xi,
                         const float* __restrict__ rg) {
  int i = blockIdx.x * blockDim.x + threadIdx.x;
  if (i < TOTAL) { float r = rg[i / FN]; xr[i] *= r; xi[i] *= r; }
}

__global__ void k_scaleW(float* __restrict__ wr, float* __restrict__ wi,
                         const float* __restrict__ rg) {
  int i = blockIdx.x * blockDim.x + threadIdx.x;
  if (i < WTOT) { float r = rg[i / (NF * NCH)]; wr[i] *= r; wi[i] *= r; }
}

// -------- frame reduction via WMMA f32 16x16x4 (B = ones => row sums) --------
// rows M = 16 frequencies of a tile, K = 4 frames/step. A layout: lane l -> row
// l&15, supplies K = 2*(l>>4)+{0,1}. D: lanes with (l&15)==0 hold rows j+8*(l>>4).
// Frames are split into NSPLIT chunks (blockIdx.y encodes chunk * NCH + c);
// per-chunk partial sums are written to part-buffers summed by k_finalize.
__global__ __launch_bounds__(32)
void k_reduce(const float* __restrict__ xr, const float* __restrict__ xi,
              const float* __restrict__ sr_b, const float* __restrict__ si_b,
              const float* __restrict__ wgt,
              float* __restrict__ numr, float* __restrict__ numi, float* __restrict__ den,
              int src, int shift) {
  int tile = blockIdx.x;
  int c = blockIdx.y % NCH;
  int chunk = blockIdx.y / NCH;
  int b = blockIdx.z;
  int lane = threadIdx.x;
  int m = lane & 15, hi = lane >> 4;
  int f = tile * 16 + m;
  int fc = (f < NF) ? f : (NF - 1);
  const float* pxr = xr + (size_t)((b * NCH + c) * NF + fc) * NN;
  const float* pxi = xi + (size_t)((b * NCH + c) * NF + fc) * NN;
  const float* psr = sr_b + (size_t)((b * NCH + src) * NF + fc) * NN;
  const float* psi = si_b + (size_t)((b * NCH + src) * NF + fc) * NN;
  const float* pw = wgt + (size_t)(b * NCH + c) * NN;

  v8f anr = {}, ani = {}, adn = {};
  v2f ones = {1.f, 1.f};
  int base = 2 * hi;
  int n0 = chunk * NNCHUNK;
  int nEnd = n0 + NNCHUNK;

  if (chunk == 0) {
    // peeled first 4-frame group: only place where zero-padding (shift<=3) can
    // hit; branchless clamped loads + value masks keep EXEC uniform.
    int k0 = base, k1 = base + 1;
    float w0 = pw[k0], w1 = pw[k1];
    float xr0 = pxr[k0], xr1 = pxr[k1];
    float xi0 = pxi[k0], xi1 = pxi[k1];
    int p0 = k0 - shift, p1 = k1 - shift;
    int q0 = (p0 < 0) ? 0 : p0;
    int q1 = (p1 < 0) ? 0 : p1;
    float m0 = (p0 < 0) ? 0.f : 1.f;
    float m1 = (p1 < 0) ? 0.f : 1.f;
    float sr0 = psr[q0] * m0, si0 = psi[q0] * m0;
    float sr1 = psr[q1] * m1, si1 = psi[q1] * m1;

    v2f a;
    a.x = w0 * (xr0 * sr0 + xi0 * si0);
    a.y = w1 * (xr1 * sr1 + xi1 * si1);
    anr = __builtin_amdgcn_wmma_f32_16x16x4_f32(false, a, false, ones, (short)0, anr, false, false);
    a.x = w0 * (xi0 * sr0 - xr0 * si0);
    a.y = w1 * (xi1 * sr1 - xr1 * si1);
    ani = __builtin_amdgcn_wmma_f32_16x16x4_f32(false, a, false, ones, (short)0, ani, false, false);
    a.x = w0 * (sr0 * sr0 + si0 * si0);
    a.y = w1 * (sr1 * sr1 + si1 * si1);
    adn = __builtin_amdgcn_wmma_f32_16x16x4_f32(false, a, false, ones, (short)0, adn, false, false);
    n0 = 4;
  }

  // main loop: k - shift >= 4 - 3 >= 1, so all loads unconditional.
  for (; n0 < nEnd; n0 += 4) {
    int k0 = n0 + base, k1 = k0 + 1;
    float w0 = pw[k0], w1 = pw[k1];
    float xr0 = pxr[k0], xr1 = pxr[k1];
    float xi0 = pxi[k0], xi1 = pxi[k1];
    float sr0 = psr[k0 - shift], si0 = psi[k0 - shift];
    float sr1 = psr[k1 - shift], si1 = psi[k1 - shift];

    v2f a;
    // num_re: w * (Xr*Sr + Xi*Si)   (X * conj(S))
    a.x = w0 * (xr0 * sr0 + xi0 * si0);
    a.y = w1 * (xr1 * sr1 + xi1 * si1);
    anr = __builtin_amdgcn_wmma_f32_16x16x4_f32(false, a, false, ones, (short)0, anr, false, false);
    // num_im: w * (Xi*Sr - Xr*Si)
    a.x = w0 * (xi0 * sr0 - xr0 * si0);
    a.y = w1 * (xi1 * sr1 - xr1 * si1);
    ani = __builtin_amdgcn_wmma_f32_16x16x4_f32(false, a, false, ones, (short)0, ani, false, false);
    // denom: w * |S|^2
    a.x = w0 * (sr0 * sr0 + si0 * si0);
    a.y = w1 * (sr1 * sr1 + si1 * si1);
    adn = __builtin_amdgcn_wmma_f32_16x16x4_f32(false, a, false, ones, (short)0, adn, false, false);
  }

  if (m == 0) {  // lanes 0 and 16: rows j + 8*hi, all columns identical
    for (int j = 0; j < 8; ++j) {
      int fo = tile * 16 + j + 8 * hi;
      if (fo < NF) {
        int o = chunk * BCF + (b * NCH + c) * NF + fo;
        numr[o] = anr[j];
        numi[o] = ani[j];
        den[o] = adn[j];
      }
    }
  }
}

// -------- sum partials; v = num/max(den,eps) (with norm); type1 -> v_s on src row --------
__global__ void k_finalize(const float* __restrict__ numr, const float* __restrict__ numi,
                           const float* __restrict__ den, float* __restrict__ vr,
                           float* __restrict__ vi, int src, int type1) {
  int i = blockIdx.x * blockDim.x + threadIdx.x;
  if (i >= BCF) return;
  int c = (i / NF) % NCH;
  float nr = 0.f, ni = 0.f, dd = 0.f;
#pragma unroll
  for (int ch = 0; ch < NSPLIT; ++ch) {
    int o = ch * BCF + i;
    nr += numr[o];
    ni += numi[o];
    dd += den[o];
  }
  const float norm = 1.f / (float)NN;
  nr *= norm; ni *= norm; dd *= norm;
  float dm = fmaxf(dd, EPSV);
  float r = nr / dm, im = ni / dm;
  if (type1 && c == src) {
    r = 1.f - 1.f / sqrtf(fmaxf(dd, EPSV));  // dd == den[src] since c==src
    im = 0.f;
  }
  vr[i] = r;
  vi[i] = im;
}

// -------- X[b,c,f,:] -= v[b,c,f] * X[b,src,f,:] (old src row) --------
__global__ void k_applyX(float* __restrict__ xr, float* __restrict__ xi,
                         const float* __restrict__ vr, const float* __restrict__ vi, int src) {
  int i = blockIdx.x * blockDim.x + threadIdx.x;
  if (i >= BATCH * FN) return;
  int b = i / FN;
  int fn = i % FN;
  int f = fn / NN;
  size_t base = (size_t)b * CFN + fn;
  size_t sidx = base + (size_t)src * FN;
  float sr = xr[sidx], si = xi[sidx];
  int vb = b * NCH * NF + f;
#pragma unroll
  for (int c = 0; c < NCH; ++c) {
    float cr = vr[vb + c * NF], ci = vi[vb + c * NF];
    size_t o = base + (size_t)c * FN;
    float nr = xr[o] - (cr * sr - ci * si);
    float ni = xi[o] - (cr * si + ci * sr);
    xr[o] = nr;
    xi[o] = ni;
  }
}

// -------- type-2: X -= v * Xbar (original input delayed by `shift`) --------
__global__ void k_applyX2(float* __restrict__ xr, float* __restrict__ xi,
                          const float* __restrict__ inr, const float* __restrict__ ini,
                          const float* __restrict__ vr, const float* __restrict__ vi,
                          int src, int shift) {
  int i = blockIdx.x * blockDim.x + threadIdx.x;
  if (i >= BATCH * FN) return;
  int b = i / FN;
  int fn = i % FN;
  int f = fn / NN;
  int n = fn % NN;
  int p = n - shift;
  int q = (p < 0) ? 0 : p;
  float mk = (p < 0) ? 0.f : 1.f;
  size_t sb = (size_t)(b * NCH + src) * FN + (size_t)f * NN;
  float sr = inr[sb + q] * mk;
  float si = ini[sb + q] * mk;
  size_t base = (size_t)b * CFN + fn;
  int vb = b * NCH * NF + f;
#pragma unroll
  for (int c = 0; c < NCH; ++c) {
    float cr = vr[vb + c * NF], ci = vi[vb + c * NF];
    size_t o = base + (size_t)c * FN;
    float nr = xr[o] - (cr * sr - ci * si);
    float ni = xi[o] - (cr * si + ci * sr);
    xr[o] = nr;
    xi[o] = ni;
  }
}

// -------- W[b,c,f,:] -= v[b,c,f] * W[b,src,f,:] --------
__global__ void k_applyW(float* __restrict__ wr, float* __restrict__ wi,
                         const float* __restrict__ vr, const float* __restrict__ vi, int src) {
  int i = blockIdx.x * blockDim.x + threadIdx.x;
  if (i >= BATCH * NF * NCH) return;
  int b = i / (NF * NCH);
  int rem = i % (NF * NCH);
  int f = rem / NCH, d = rem % NCH;
  size_t base = ((size_t)(b * NCH) * NF + f) * NCH + d;
  size_t sidx = base + (size_t)src * NF * NCH;
  float sr = wr[sidx], si = wi[sidx];
  int vb = b * NCH * NF + f;
#pragma unroll
  for (int c = 0; c < NCH; ++c) {
    float cr = vr[vb + c * NF], ci = vi[vb + c * NF];
    size_t o = base + (size_t)c * NF * NCH;
    float nr = wr[o] - (cr * sr - ci * si);
    float ni = wi[o] - (cr * si + ci * sr);
    wr[o] = nr;
    wi[o] = ni;
  }
}

// -------- projection back: solve (W^T + eps I) a = e1 per (b,f) --------
__global__ void k_proj(const float* __restrict__ wr, const float* __restrict__ wi,
                       float* __restrict__ ar_, float* __restrict__ ai_) {
  int i = blockIdx.x * blockDim.x + threadIdx.x;
  if (i >= BATCH * NF) return;
  int b = i / NF, f = i % NF;
  float Ar[4][5], Ai[4][5];
  for (int d = 0; d < 4; ++d) {
    for (int c = 0; c < 4; ++c) {
      size_t o = ((size_t)(b * NCH + c) * NF + f) * NCH + d;
      Ar[d][c] = wr[o];
      Ai[d][c] = wi[o];
    }
    Ar[d][d] += EPSV;
    Ar[d][4] = (d == 0) ? 1.f : 0.f;
    Ai[d][4] = 0.f;
  }
  for (int col = 0; col < 4; ++col) {
    int piv = col;
    float best = Ar[col][col] * Ar[col][col] + Ai[col][col] * Ai[col][col];
    for (int r = col + 1; r < 4; ++r) {
      float mg = Ar[r][col] * Ar[r][col] + Ai[r][col] * Ai[r][col];
      if (mg > best) { best = mg; piv = r; }
    }
    if (piv != col)
      for (int k = 0; k < 5; ++k) {
        float t = Ar[col][k]; Ar[col][k] = Ar[piv][k]; Ar[piv][k] = t;
        t = Ai[col][k]; Ai[col][k] = Ai[piv][k]; Ai[piv][k] = t;
      }
    float pr = Ar[col][col], pq = Ai[col][col];
    float inv = 1.f / fmaxf(pr * pr + pq * pq, 1e-30f);
    float qr = pr * inv, qi = -pq * inv;
    for (int k = 0; k < 5; ++k) {
      float tr = Ar[col][k], ti = Ai[col][k];
      Ar[col][k] = tr * qr - ti * qi;
      Ai[col][k] = tr * qi + ti * qr;
    }
    for (int r = 0; r < 4; ++r) {
      if (r == col) continue;
      float fr = Ar[r][col], fi = Ai[r][col];
      for (int k = 0; k < 5; ++k) {
        float tr = Ar[col][k], ti = Ai[col][k];
        Ar[r][k] -= fr * tr - fi * ti;
        Ai[r][k] -= fr * ti + fi * tr;
      }
    }
  }
  for (int c = 0; c < 4; ++c) {
    int o = (b * NCH + c) * NF + f;
    ar_[o] = Ar[c][4];
    ai_[o] = Ai[c][4];
  }
}

__global__ void k_applya(float* __restrict__ xr, float* __restrict__ xi,
                         const float* __restrict__ ar_, const float* __restrict__ ai_) {
  int i = blockIdx.x * blockDim.x + threadIdx.x;
  if (i >= TOTAL) return;
  int bc = i / FN;
  int f = (i % FN) / NN;
  int o = bc * NF + f;
  float r = ar_[o], im = ai_[o];
  float a = xr[i], b = xi[i];
  xr[i] = a * r - b * im;
  xi[i] = a * im + b * r;
}

extern "C" void kernel_launch(void* const* d_in, const int* in_sizes, int n_in,
                              void* d_out, int out_size, void* d_ws, size_t ws_size,
                              hipStream_t stream) {
  (void)in_sizes; (void)n_in; (void)out_size; (void)ws_size;
  const float* inr = (const float*)d_in[0];
  const float* ini = (const float*)d_in[1];
  float* xr = (float*)d_out;        // output layout: [re plane | im plane]
  float* xi = xr + TOTAL;

  float* ws = (float*)d_ws;
  float* s    = ws; ws += BCN;
  float* wgt  = ws; ws += BCN;
  float* g    = ws; ws += 16;
  float* rg   = ws; ws += 16;
  float* numr = ws; ws += BCF * NSPLIT;
  float* numi = ws; ws += BCF * NSPLIT;
  float* den  = ws; ws += BCF * NSPLIT;
  float* vr   = ws; ws += BCF;
  float* vi   = ws; ws += BCF;
  float* Wr   = ws; ws += WTOT;
  float* Wi   = ws; ws += WTOT;
  float* ar_  = ws; ws += BCF;
  float* ai_  = ws; ws += BCF;

  dim3 blk(256);
  k_init_copy<<<dim3((TOTAL + 255) / 256), blk, 0, stream>>>(inr, ini, xr, xi);
  k_init_W<<<dim3((WTOT + 255) / 256), blk, 0, stream>>>(Wr, Wi);

  dim3 rgrid((NF + 15) / 16, NCH * NSPLIT, BATCH);  // 17 x 20 x 4 one-wave blocks
  for (int it = 0; it < NITER; ++it) {
    k_colsum<<<dim3((NN + 255) / 256, NCH, BATCH), blk, 0, stream>>>(xr, xi, s);
    k_gweights<<<dim3(BATCH * NCH), blk, 0, stream>>>(s, wgt, g, rg);
    k_scaleX<<<dim3((TOTAL + 255) / 256), blk, 0, stream>>>(xr, xi, rg);
    k_scaleW<<<dim3((WTOT + 255) / 256), blk, 0, stream>>>(Wr, Wi, rg);

    for (int src = 0; src < NCH; ++src) {
      k_reduce<<<rgrid, dim3(32), 0, stream>>>(xr, xi, xr, xi, wgt, numr, numi, den, src, 0);
      k_finalize<<<dim3((BCF + 255) / 256), blk, 0, stream>>>(numr, numi, den, vr, vi, src, 1);
      k_applyX<<<dim3((BATCH * FN + 255) / 256), blk, 0, stream>>>(xr, xi, vr, vi, src);
      k_applyW<<<dim3((BATCH * NF * NCH + 255) / 256), blk, 0, stream>>>(Wr, Wi, vr, vi, src);
    }
    for (int src = 0; src < NCH; ++src) {
      for (int tap = 0; tap < NTAPS; ++tap) {
        int shift = PTOT - tap;  // X_bar tap t == input delayed by 3 - t
        k_reduce<<<rgrid, dim3(32), 0, stream>>>(xr, xi, inr, ini, wgt, numr, numi, den, src, shift);
        k_finalize<<<dim3((BCF + 255) / 256), blk, 0, stream>>>(numr, numi, den, vr, vi, src, 0);
        k_applyX2<<<dim3((BATCH * FN + 255) / 256), blk, 0, stream>>>(xr, xi, inr, ini, vr, vi, src, shift);
      }
    }
  }
  k_proj<<<dim3((BATCH * NF + 255) / 256), blk, 0, stream>>>(Wr, Wi, ar_, ai_);
  k_applya<<<dim3((TOTAL + 255) / 256), blk, 0, stream>>>(xr, xi, ar_, ai_);
}